// MaskedConv2d_79748952752305
// MI455X (gfx1250) — compile-verified
//
#include <hip/hip_runtime.h>

typedef __attribute__((ext_vector_type(2))) float v2f;
typedef __attribute__((ext_vector_type(8))) float v8f;

// Per-output-channel connectivity (LeNet C3 with "last h wins" overwrite):
//  h_sel[n]-2 packed as 2-bit fields, k_sel[n] packed as 4-bit fields.
//  h_sel = {2,3,4,5,5,5,3,4,5,5,5,5,4,5,5,5}
//  k_sel = {0,0,0,0,1,2,3,3,3,4,5,6,7,7,8,9}
#define HCODE 0xFEFF9FE4u
#define KCODE 0x9877654333210000ull

#define IMGS_PER_WAVE 4

__global__ __launch_bounds__(32) void c3_wmma_kernel(
    const float* __restrict__ x, const float* __restrict__ w,
    const float* __restrict__ bias, float* __restrict__ out, int nimg)
{
    // Masked weight matrix, pair-interleaved: BI[(t*2+pp)*16+n] = {B[t,2pp][n], B[t,2pp+1][n]}
    __shared__ v2f ldsBI[800];          //  6400 B
    // Input, pair-interleaved: XI[img*392 + pp*196 + pixel] = {ch(2+2pp)[pix], ch(3+2pp)[pix]}
    __shared__ v2f ldsXI[4 * 392];      // 12544 B
    // Output staging, per image channel-major (1600 floats/image)
    __shared__ float ldsO[4 * 1600];    // 25600 B

    const int lane = threadIdx.x;                 // wave32: 0..31
    const int base = blockIdx.x * IMGS_PER_WAVE;  // first image of this wave
    if (base >= nimg) return;
    const int cnt = (nimg - base < IMGS_PER_WAVE) ? (nimg - base) : IMGS_PER_WAVE;

    // ---- Build masked 100x16 weight matrix as v2f pairs (zero-padded GEMM-K) ----
    for (int e = lane; e < 800; e += 32) {
        int n = e & 15, tp = e >> 4, t = tp >> 1, pp = tp & 1;
        int h = 2 + (int)((HCODE >> (2 * n)) & 3u);
        int k = (int)((KCODE >> (4 * n)) & 15ull);
        float wv = w[(h * 10 + k) * 25 + t];      // weights[h][k][t/5][t%5]
        v2f v;
        v.x = (h == 2 * pp + 2) ? wv : 0.0f;
        v.y = (h == 2 * pp + 3) ? wv : 0.0f;
        ldsBI[e] = v;
    }

    // ---- Stage input channels 2..5, pair-interleaved (coalesced b128 reads) ----
    for (int img = 0; img < cnt; ++img) {
        const float* xb = x + (size_t)(base + img) * 1176 + 392;  // ch2 plane
        #pragma unroll
        for (int pr = 0; pr < 2; ++pr) {
            const float4* sA = (const float4*)(xb + pr * 392);        // ch 2+2pr
            const float4* sB = (const float4*)(xb + pr * 392 + 196);  // ch 3+2pr
            v2f* dst = &ldsXI[img * 392 + pr * 196];
            for (int r = lane; r < 49; r += 32) {                     // 49 quads
                float4 a4 = sA[r], b4 = sB[r];
                v2f p0; p0.x = a4.x; p0.y = b4.x;
                v2f p1; p1.x = a4.y; p1.y = b4.y;
                v2f p2; p2.x = a4.z; p2.y = b4.z;
                v2f p3; p3.x = a4.w; p3.y = b4.w;
                dst[r * 4 + 0] = p0;
                dst[r * 4 + 1] = p1;
                dst[r * 4 + 2] = p2;
                dst[r * 4 + 3] = p3;
            }
        }
    }
    __syncthreads();

    const int n  = lane & 15;        // N index (output channel)
    const int hi = lane >> 4;        // lane half -> K pair {0,1} or {2,3}

    // ---- Preload all 25 B fragments (single b64 each), reused by all M-tiles ----
    v2f bf[25];
    #pragma unroll
    for (int t = 0; t < 25; ++t) bf[t] = ldsBI[(t * 2 + hi) * 16 + n];

    const float bv = bias[n];

    // ---- 25 M-tiles of 16 pixels (4 images * 100 px = 400 = 25*16, no padding) ----
    #pragma unroll 1
    for (int tile = 0; tile < 25; ++tile) {
        int m  = tile * 16 + n;          // flat pixel id over the 4 images
        int im = m / 100;
        int p  = m - im * 100;
        int yy = p / 10;
        int xx = p - yy * 10;
        const v2f* ap = &ldsXI[im * 392 + hi * 196 + yy * 14 + xx];

        v8f acc;
        #pragma unroll
        for (int q = 0; q < 8; ++q) acc[q] = bv;   // seed with bias

        #pragma unroll
        for (int t = 0; t < 25; ++t) {             // 25 K-slices of 4
            const int off = (t / 5) * 14 + (t % 5);
            acc = __builtin_amdgcn_wmma_f32_16x16x4_f32(
                false, ap[off], false, bf[t], (short)0, acc, false, false);
        }

        // ---- Transpose D through LDS (C/D layout: M = q + 8*half, N = lane&15) ----
        int mb = tile * 16 + hi * 8;
        #pragma unroll
        for (int q = 0; q < 8; ++q) {
            int mm = mb + q;
            int ii = mm / 100;
            int pp = mm - ii * 100;
            ldsO[ii * 1600 + n * 100 + pp] = acc[q];
        }
    }

    __syncthreads();  // DS-ordering before reading staged output

    // ---- Coalesced b128 store: each image's output is 1600 contiguous floats ----
    float4* go = (float4*)(out + (size_t)base * 1600);
    const float4* lo = (const float4*)ldsO;
    const int n4 = cnt * 400;
    for (int r = lane; r < n4; r += 32) go[r] = lo[r];
}

extern "C" void kernel_launch(void* const* d_in, const int* in_sizes, int n_in,
                              void* d_out, int out_size, void* d_ws, size_t ws_size,
                              hipStream_t stream) {
    const float* x    = (const float*)d_in[0];   // [B, 6, 14, 14] fp32
    const float* w    = (const float*)d_in[1];   // [6, 10, 5, 5]  fp32
    const float* bias = (const float*)d_in[2];   // [1, 16, 1, 1]  fp32
    float* out = (float*)d_out;                  // [B, 16, 10, 10] fp32

    const int nimg   = in_sizes[0] / (6 * 14 * 14);
    const int blocks = (nimg + IMGS_PER_WAVE - 1) / IMGS_PER_WAVE;
    hipLaunchKernelGGL(c3_wmma_kernel, dim3(blocks), dim3(32), 0, stream,
                       x, w, bias, out, nimg);
}